// PulseGAT_82729660055640
// MI455X (gfx1250) — compile-verified
//
#include <hip/hip_runtime.h>
#include <math.h>

typedef __attribute__((ext_vector_type(2))) float v2f;
typedef __attribute__((ext_vector_type(8))) float v8f;
typedef __attribute__((ext_vector_type(4))) unsigned int u32x4;
typedef __attribute__((ext_vector_type(8))) int i32x8;
typedef __attribute__((ext_vector_type(4))) int i32x4;

#define HEADS 4
#define HID 64
#define IN_C 64
#define NUM_GRAPHS 64
#define NEG_SLOPE 0.2f

#define KCHUNK 64                      // K rows staged in LDS per TDM round
#define A_STRIDE (KCHUNK + 2)          // padded LDS row stride for A (bank-conflict free)

// ---------------------------------------------------------------------------
// Tensor Data Mover: issue a 2-D tile load (global -> LDS).
// D# bitfields per CDNA5 ISA 8.3/8.4. This toolchain exposes the 6-arg
// builtin (g0 u32x4, g1 i32x8, g2 i32x4, g3 i32x4, g4 i32x8, cpol); groups
// beyond g1 are zero for a 2-D tensor. data_size = 4 bytes (code 2). type=2.
// ---------------------------------------------------------------------------
__device__ __forceinline__ void tdm_load_2d(const void* gaddr, unsigned lds_off,
                                            unsigned tensor_d0, unsigned tensor_d1,
                                            unsigned tile_d0, unsigned tile_d1,
                                            unsigned stride0,
                                            int pad_en, unsigned pad_interval_code,
                                            unsigned pad_amount_code) {
    unsigned long long ga = (unsigned long long)gaddr;
    u32x4 g0;
    g0.x = 1u;                                   // count=1, user descriptor
    g0.y = lds_off;                              // lds_addr (bytes)
    g0.z = (unsigned)(ga & 0xFFFFFFFFu);         // global_addr[31:0]
    g0.w = (unsigned)((ga >> 32) & 0x01FFFFFFu)  // global_addr[56:32]
         | 0x80000000u;                          // type=2 at bits[127:126]
    unsigned w0 = (2u << 16);                    // data_size = 4B
    if (pad_en) w0 |= (1u << 20) | (pad_interval_code << 22) | (pad_amount_code << 25);
    i32x8 g1;
    g1[0] = (int)w0;
    g1[1] = (int)((tensor_d0 & 0xFFFFu) << 16);                       // dim0[15:0] @63:48
    g1[2] = (int)(((tensor_d0 >> 16) & 0xFFFFu) | ((tensor_d1 & 0xFFFFu) << 16));
    g1[3] = (int)(((tensor_d1 >> 16) & 0xFFFFu) | ((tile_d0 & 0xFFFFu) << 16));
    g1[4] = (int)(tile_d1 & 0xFFFFu);                                 // tile_dim1; tile_dim2=0
    g1[5] = (int)stride0;                                             // dim0_stride[31:0]
    g1[6] = 0;                                                        // stride hi, dim1_stride lo
    g1[7] = 0;
    i32x4 g2 = {0, 0, 0, 0};
    i32x4 g3 = {0, 0, 0, 0};
    i32x8 g4 = {0, 0, 0, 0, 0, 0, 0, 0};
    __builtin_amdgcn_tensor_load_to_lds(g0, g1, g2, g3, g4, 0);
}

// ---------------------------------------------------------------------------
// fp32 WMMA GEMM with TDM->LDS staging.
// Block = 128 threads (4 wave32s): one 16-row tile x 64 columns; wave w owns
// col tile tn_base + 16*w. K staged in 64-deep chunks: wave 0 TDM-loads the
// B chunk (KCHUNK x 64, row-major) and the A chunk (16 x KCHUNK, padded row
// stride A_STRIDE) into LDS, waits TENSORcnt, block barriers, then all waves
// run V_WMMA_F32_16X16X4_F32 off LDS.
// Requires M%16==0, N%64==0, K%KCHUNK==0 (holds: M=50000, N in {64,256},
// K in {64,256}).
// ---------------------------------------------------------------------------
__global__ void gemm_f32_wmma_tdm(const float* __restrict__ A,
                                  const float* __restrict__ B,
                                  float* __restrict__ C,
                                  int M, int N, int K) {
    extern __shared__ float smem[];              // [KCHUNK*64] B ++ [16*A_STRIDE] A
    float* smemB = smem;
    float* smemA = smem + KCHUNK * 64;

    const int colGroups = N >> 6;                // 64 columns per block
    const int tm      = (blockIdx.x / colGroups) << 4;
    const int tn_base = (blockIdx.x % colGroups) << 6;
    const int wave    = threadIdx.x >> 5;        // 0..3
    const int lane    = threadIdx.x & 31;
    const int half    = lane >> 4;
    const int l16     = lane & 15;
    const int tn_loc  = wave << 4;               // column tile inside LDS chunk

    const unsigned ldsB = (unsigned)(uintptr_t)(void*)smemB;  // LDS byte offsets
    const unsigned ldsA = (unsigned)(uintptr_t)(void*)smemA;

    v8f acc = {};
    for (int kc = 0; kc < K; kc += KCHUNK) {
        if (threadIdx.x < 32) {
            // B chunk: rows kc..kc+KCHUNK-1, cols tn_base..tn_base+63
            tdm_load_2d(B + (size_t)kc * N + tn_base, ldsB,
                        (unsigned)N, (unsigned)(K - kc),
                        64u, (unsigned)KCHUNK, (unsigned)N,
                        /*pad=*/0, 0u, 0u);
            // A chunk: rows tm..tm+15, cols kc..kc+KCHUNK-1; pad 2 DWORDs per
            // 64 DWORDs (interval code 5, amount code 1) -> LDS stride 66
            tdm_load_2d(A + (size_t)tm * K + kc, ldsA,
                        (unsigned)K, (unsigned)(M - tm),
                        (unsigned)KCHUNK, 16u, (unsigned)K,
                        /*pad=*/1, 5u, 1u);
            __builtin_amdgcn_s_wait_tensorcnt(0);
        }
        __syncthreads();

#pragma unroll
        for (int k0 = 0; k0 < KCHUNK; k0 += 4) {
            const int ka = k0 + (half << 1);
            v2f a, b;
            a.x = smemA[l16 * A_STRIDE + ka];
            a.y = smemA[l16 * A_STRIDE + ka + 1];
            b.x = smemB[ka * 64 + tn_loc + l16];
            b.y = smemB[(ka + 1) * 64 + tn_loc + l16];
            acc = __builtin_amdgcn_wmma_f32_16x16x4_f32(
                false, a, false, b, (short)0, acc, false, false);
        }
        __syncthreads();
    }

#pragma unroll
    for (int r = 0; r < 8; ++r) {
        C[(size_t)(tm + r + (half << 3)) * N + (tn_base + tn_loc + l16)] = acc[r];
    }
}

// ---------------------------------------------------------------------------
// Attention score dot products: es[n,h] = <xp[n,h,:], a_src[h,:]>, same for ed
// ---------------------------------------------------------------------------
__global__ void att_scores(const float* __restrict__ xp,
                           const float* __restrict__ a_src,
                           const float* __restrict__ a_dst,
                           float* __restrict__ es, float* __restrict__ ed,
                           int n, int H, int C) {
    int i = blockIdx.x * blockDim.x + threadIdx.x;
    if (i >= n * H) return;
    int node = i / H, h = i % H;
    const float* xr = xp + (size_t)node * H * C + (size_t)h * C;
    const float* as = a_src + (size_t)h * C;
    const float* ad = a_dst + (size_t)h * C;
    float s = 0.f, d = 0.f;
    for (int c = 0; c < C; ++c) { s += xr[c] * as[c]; d += xr[c] * ad[c]; }
    es[i] = s;
    ed[i] = d;
}

__global__ void fill_f32(float* __restrict__ p, float v, size_t n) {
    size_t i = (size_t)blockIdx.x * blockDim.x + threadIdx.x;
    if (i < n) p[i] = v;
}

__device__ __forceinline__ float lrelu(float x) {
    return x > 0.f ? x : NEG_SLOPE * x;
}

// sign-aware float atomic max via monotone integer ordering
__device__ __forceinline__ void atomicMaxF(float* addr, float val) {
    if (val >= 0.f) atomicMax((int*)addr, __float_as_int(val));
    else            atomicMin((unsigned int*)addr, __float_as_uint(val));
}

// ---------------------------------------------------------------------------
// Segment softmax pass 1: per-dst running max of leaky_relu(es[src]+ed[dst])
// Edges e < E come from edge_index; e >= E are self loops (src=dst=e-E).
// ---------------------------------------------------------------------------
__global__ void edge_max(const int* __restrict__ srcs, const int* __restrict__ dsts,
                         int E, int n,
                         const float* __restrict__ es, const float* __restrict__ ed,
                         float* __restrict__ m, int H) {
    int e = blockIdx.x * blockDim.x + threadIdx.x;
    int ET = E + n;
    if (e >= ET) return;
    int s = (e < E) ? srcs[e] : (e - E);
    int d = (e < E) ? dsts[e] : (e - E);
    for (int h = 0; h < H; ++h) {
        float v = lrelu(es[s * H + h] + ed[d * H + h]);
        atomicMaxF(&m[d * H + h], v);
    }
}

// Segment softmax pass 2: denom[dst,h] += exp(e - max)
__global__ void edge_den(const int* __restrict__ srcs, const int* __restrict__ dsts,
                         int E, int n,
                         const float* __restrict__ es, const float* __restrict__ ed,
                         const float* __restrict__ m, float* __restrict__ den, int H) {
    int e = blockIdx.x * blockDim.x + threadIdx.x;
    int ET = E + n;
    if (e >= ET) return;
    int s = (e < E) ? srcs[e] : (e - E);
    int d = (e < E) ? dsts[e] : (e - E);
    for (int h = 0; h < H; ++h) {
        float v = lrelu(es[s * H + h] + ed[d * H + h]);
        atomicAdd(&den[d * H + h], __expf(v - m[d * H + h]));
    }
}

// Pass 3: out[dst,h,c] += alpha(e,h) * xp[src,h,c].  One block per edge,
// blockDim = H*C threads (256 for conv1, 64 for conv2).
__global__ void edge_agg(const int* __restrict__ srcs, const int* __restrict__ dsts,
                         int E, int n,
                         const float* __restrict__ es, const float* __restrict__ ed,
                         const float* __restrict__ m, const float* __restrict__ den,
                         const float* __restrict__ xp, float* __restrict__ out,
                         int H, int C) {
    int e = blockIdx.x;
    int t = threadIdx.x;          // 0..H*C-1 ; h = t/C, c = t%C
    int h = t / C;
    int s = (e < E) ? srcs[e] : (e - E);
    int d = (e < E) ? dsts[e] : (e - E);
    float v     = lrelu(es[s * H + h] + ed[d * H + h]);
    float alpha = __expf(v - m[d * H + h]) / den[d * H + h];
    atomicAdd(&out[(size_t)d * H * C + t], alpha * xp[(size_t)s * H * C + t]);
}

__global__ void bias_elu(float* __restrict__ h, const float* __restrict__ b,
                         int n, int F, int apply_elu) {
    int i = blockIdx.x * blockDim.x + threadIdx.x;
    if (i >= n * F) return;
    float v = h[i] + b[i % F];
    if (apply_elu) v = (v > 0.f) ? v : (__expf(v) - 1.f);
    h[i] = v;
}

// global mean pool, accumulation phase
__global__ void pool_sum(const float* __restrict__ h2, const int* __restrict__ batch,
                         float* __restrict__ pooled, float* __restrict__ cnt,
                         int n, int F) {
    int i = blockIdx.x * blockDim.x + threadIdx.x;
    if (i >= n * F) return;
    int node = i / F, c = i % F;
    int g = batch[node];
    atomicAdd(&pooled[g * F + c], h2[i]);
    if (c == 0) atomicAdd(&cnt[g], 1.f);
}

// final MLP head: one thread per graph (64 graphs)
__global__ void mlp_head(const float* __restrict__ pooled, const float* __restrict__ cnt,
                         const float* __restrict__ Wh1, const float* __restrict__ bh1,
                         const float* __restrict__ Wh2, const float* __restrict__ bh2,
                         float* __restrict__ out) {
    int g = threadIdx.x;
    if (g >= NUM_GRAPHS) return;
    float inv = 1.f / fmaxf(cnt[g], 1.f);
    float z[16];
#pragma unroll
    for (int j = 0; j < 16; ++j) {
        float acc = bh1[j];
        for (int c = 0; c < HID; ++c) acc += pooled[g * HID + c] * inv * Wh1[c * 16 + j];
        z[j] = fmaxf(acc, 0.f);
    }
    float o = bh2[0];
#pragma unroll
    for (int j = 0; j < 16; ++j) o += z[j] * Wh2[j];
    out[g] = 1.f / (1.f + __expf(-o));
}

extern "C" void kernel_launch(void* const* d_in, const int* in_sizes, int n_in,
                              void* d_out, int out_size, void* d_ws, size_t ws_size,
                              hipStream_t stream) {
    const float* x      = (const float*)d_in[0];
    const int*   eidx   = (const int*)d_in[1];     // [2,E] flattened
    const int*   batch  = (const int*)d_in[2];
    const float* W1     = (const float*)d_in[3];
    const float* a1s    = (const float*)d_in[4];
    const float* a1d    = (const float*)d_in[5];
    const float* b1     = (const float*)d_in[6];
    const float* W2     = (const float*)d_in[7];
    const float* a2s    = (const float*)d_in[8];
    const float* a2d    = (const float*)d_in[9];
    const float* b2     = (const float*)d_in[10];
    const float* Wh1    = (const float*)d_in[11];
    const float* bh1    = (const float*)d_in[12];
    const float* Wh2    = (const float*)d_in[13];
    const float* bh2    = (const float*)d_in[14];
    float* out = (float*)d_out;

    const int N  = in_sizes[2];          // 50000 nodes
    const int E  = in_sizes[1] / 2;      // 800000 edges
    const int ET = E + N;                // edges + self loops
    const int* srcs = eidx;
    const int* dsts = eidx + E;
    const int F1 = HEADS * HID;          // 256

    // workspace layout (floats)
    float* ws = (float*)d_ws;
    size_t o = 0;
    float* xp1  = ws + o; o += (size_t)N * F1;   // conv1 projected features
    float* es1  = ws + o; o += (size_t)N * HEADS;
    float* ed1  = ws + o; o += (size_t)N * HEADS;
    float* m1   = ws + o; o += (size_t)N * HEADS;
    float* den1 = ws + o; o += (size_t)N * HEADS;
    float* h1   = ws + o; o += (size_t)N * F1;   // conv1 aggregate -> ELU output
    float* xp2  = ws + o; o += (size_t)N * HID;
    float* es2  = ws + o; o += (size_t)N;
    float* ed2  = ws + o; o += (size_t)N;
    float* m2   = ws + o; o += (size_t)N;
    float* den2 = ws + o; o += (size_t)N;
    float* h2   = ws + o; o += (size_t)N * HID;  // conv2 aggregate output
    float* pooled = ws + o; o += (size_t)NUM_GRAPHS * HID;
    float* cnt    = ws + o; o += (size_t)NUM_GRAPHS;

    const int TB = 256;
    const float NEG_INF = -INFINITY;
    const size_t ldsBytes = (size_t)(KCHUNK * 64 + 16 * A_STRIDE) * sizeof(float);

    // ---- init scratch (every call: harness does not re-zero) ----
    {
        size_t n;
        n = (size_t)N * HEADS;
        fill_f32<<<(unsigned)((n + TB - 1) / TB), TB, 0, stream>>>(m1, NEG_INF, n);
        fill_f32<<<(unsigned)((n + TB - 1) / TB), TB, 0, stream>>>(den1, 0.f, n);
        n = (size_t)N * F1;
        fill_f32<<<(unsigned)((n + TB - 1) / TB), TB, 0, stream>>>(h1, 0.f, n);
        n = (size_t)N;
        fill_f32<<<(unsigned)((n + TB - 1) / TB), TB, 0, stream>>>(m2, NEG_INF, n);
        fill_f32<<<(unsigned)((n + TB - 1) / TB), TB, 0, stream>>>(den2, 0.f, n);
        n = (size_t)N * HID;
        fill_f32<<<(unsigned)((n + TB - 1) / TB), TB, 0, stream>>>(h2, 0.f, n);
        n = (size_t)NUM_GRAPHS * HID;
        fill_f32<<<(unsigned)((n + TB - 1) / TB), TB, 0, stream>>>(pooled, 0.f, n);
        n = (size_t)NUM_GRAPHS;
        fill_f32<<<1, TB, 0, stream>>>(cnt, 0.f, n);
    }

    // ---- conv1 ----
    // xp1 = x @ W1   (M=50000, K=64, N=256) : TDM-staged WMMA fp32
    gemm_f32_wmma_tdm<<<(N / 16) * (F1 / 64), 128, ldsBytes, stream>>>(x, W1, xp1,
                                                                       N, F1, IN_C);
    att_scores<<<(N * HEADS + TB - 1) / TB, TB, 0, stream>>>(xp1, a1s, a1d, es1, ed1,
                                                             N, HEADS, HID);
    edge_max<<<(ET + TB - 1) / TB, TB, 0, stream>>>(srcs, dsts, E, N, es1, ed1, m1, HEADS);
    edge_den<<<(ET + TB - 1) / TB, TB, 0, stream>>>(srcs, dsts, E, N, es1, ed1, m1, den1, HEADS);
    edge_agg<<<ET, HEADS * HID, 0, stream>>>(srcs, dsts, E, N, es1, ed1, m1, den1,
                                             xp1, h1, HEADS, HID);
    bias_elu<<<((size_t)N * F1 + TB - 1) / TB, TB, 0, stream>>>(h1, b1, N, F1, 1);

    // ---- conv2 (H=1, concat=False; mean over 1 head == identity) ----
    gemm_f32_wmma_tdm<<<(N / 16) * (HID / 64), 128, ldsBytes, stream>>>(h1, W2, xp2,
                                                                        N, HID, F1);
    att_scores<<<(N + TB - 1) / TB, TB, 0, stream>>>(xp2, a2s, a2d, es2, ed2, N, 1, HID);
    edge_max<<<(ET + TB - 1) / TB, TB, 0, stream>>>(srcs, dsts, E, N, es2, ed2, m2, 1);
    edge_den<<<(ET + TB - 1) / TB, TB, 0, stream>>>(srcs, dsts, E, N, es2, ed2, m2, den2, 1);
    edge_agg<<<ET, HID, 0, stream>>>(srcs, dsts, E, N, es2, ed2, m2, den2, xp2, h2, 1, HID);
    bias_elu<<<((size_t)N * HID + TB - 1) / TB, TB, 0, stream>>>(h2, b2, N, HID, 0);

    // ---- global mean pool + MLP head ----
    pool_sum<<<((size_t)N * HID + TB - 1) / TB, TB, 0, stream>>>(h2, batch, pooled, cnt, N, HID);
    mlp_head<<<1, 64, 0, stream>>>(pooled, cnt, Wh1, bh1, Wh2, bh2, out);
}